// EfficientSlidingWindowMultiheadAttention_45397804319137
// MI455X (gfx1250) — compile-verified
//
#include <hip/hip_runtime.h>
#include <hip/hip_bf16.h>
#include <math.h>

// ---------------------------------------------------------------------------
// EfficientSlidingWindowMultiheadAttention for gfx1250 (MI455X, wave32, WMMA)
// B=1, S=4096, H=1024, NUM_HEADS=16, hd=64, W=33 (read at runtime)
//
//   1) qkv = x @ W_qkv + b_qkv          (fp32 WMMA 16x16x4, async-LDS double buffered)
//   2) RoPE on q,k in-place
//   3) sliding-window attention -> ctx  (VALU; tiny FLOP count, L2-resident)
//   4) out = ctx @ W_out + b_out        (same WMMA GEMM)
// ---------------------------------------------------------------------------

typedef float v2f __attribute__((ext_vector_type(2)));
typedef float v8f __attribute__((ext_vector_type(8)));
typedef int   v4i_t __attribute__((vector_size(16)));

#define NH 16
#define HD 64
#define SEQ 4096
#define HID 1024

#define TBM 128        // block rows
#define TBN 64         // block cols
#define TBK 16         // block K
#define AS_STRIDE 20   // dword row stride for A tile in LDS (bank-safe, 16B aligned)
#define BS_STRIDE 72   // dword row stride for B tile in LDS (2*72 == 16 mod 64)

// ---- CDNA5 async global->LDS path (guarded; sync fallback keeps compile safe)
#if defined(__has_builtin)
#if __has_builtin(__builtin_amdgcn_global_load_async_to_lds_b128)
#define HAVE_ASYNC_LDS 1
#define ASYNC_COPY_B128(gsrc, ldst)                                            \
    __builtin_amdgcn_global_load_async_to_lds_b128(                            \
        (__attribute__((address_space(1))) v4i_t*)(void*)(gsrc),               \
        (__attribute__((address_space(3))) v4i_t*)(void*)(ldst), 0, 0)
#endif
#endif

#ifdef HAVE_ASYNC_LDS
#if defined(__has_builtin) && __has_builtin(__builtin_amdgcn_s_wait_asynccnt)
#define WAIT_ASYNCCNT(n) __builtin_amdgcn_s_wait_asynccnt(n)
#else
#define WAIT_ASYNCCNT(n) asm volatile("s_wait_asynccnt %0" ::"i"(n) : "memory")
#endif
#endif

// C[M,N] = A[M,K] @ B[K,N] + bias[N]   (fp32; M%128==0, N%64==0, K%16==0)
__global__ __launch_bounds__(256)
void gemm_bias_f32_wmma(const float* __restrict__ A,
                        const float* __restrict__ B,
                        const float* __restrict__ bias,
                        float* __restrict__ C,
                        int M, int N, int K) {
    __shared__ float As[2][TBM * AS_STRIDE];
    __shared__ float Bs[2][TBK * BS_STRIDE];

    const int tid  = threadIdx.x;
    const int lane = tid & 31;
    const int wave = tid >> 5;        // 0..7
    const int wm   = wave >> 1;       // 0..3 -> 32-row strip
    const int wn   = wave & 1;        // 0..1 -> 32-col strip
    const int half = lane >> 4;       // K-pair / M-half split per ISA layout
    const int l15  = lane & 15;

    const int bm = blockIdx.y * TBM;
    const int bn = blockIdx.x * TBN;

    // global->LDS mapping: A tile 128x16 (2 x b128 per thread), B tile 16x64 (1 x b128)
    const int ar   = tid >> 1;          // 0..127
    const int ac0  = (tid & 1) << 3;    // 0 or 8
    const int bkr  = tid >> 4;          // 0..15
    const int bc   = (tid & 15) << 2;   // 0..60

    const float* gA = A + (size_t)(bm + ar) * K + ac0;
    const float* gB = B + (size_t)bkr * N + bn + bc;
    const int aLds0 = ar * AS_STRIDE + ac0;
    const int bLds0 = bkr * BS_STRIDE + bc;

    auto issue = [&](int b, int kt) {
#ifdef HAVE_ASYNC_LDS
        ASYNC_COPY_B128(gA + kt,              &As[b][aLds0]);
        ASYNC_COPY_B128(gA + kt + 4,          &As[b][aLds0 + 4]);
        ASYNC_COPY_B128(gB + (size_t)kt * N,  &Bs[b][bLds0]);
#else
        const float4 av0 = *(const float4*)(gA + kt);
        const float4 av1 = *(const float4*)(gA + kt + 4);
        const float4 bv  = *(const float4*)(gB + (size_t)kt * N);
        *(float4*)&As[b][aLds0]     = av0;
        *(float4*)&As[b][aLds0 + 4] = av1;
        *(float4*)&Bs[b][bLds0]     = bv;
#endif
    };

    v8f acc00 = {}, acc01 = {}, acc10 = {}, acc11 = {};

    const int arow0 = (wm * 32 + l15) * AS_STRIDE;
    const int arow1 = arow0 + 16 * AS_STRIDE;
    const int cb    = wn * 32 + l15;

    issue(0, 0);
    int buf = 0;
    for (int kt = 0; kt < K; kt += TBK) {
        if (kt + TBK < K) {
            issue(buf ^ 1, kt + TBK);     // prefetch next tile into other buffer
#ifdef HAVE_ASYNC_LDS
            WAIT_ASYNCCNT(3);             // current tile's 3 copies complete (in-order)
#endif
        } else {
#ifdef HAVE_ASYNC_LDS
            WAIT_ASYNCCNT(0);
#endif
        }
        __syncthreads();                  // all waves' tile-i data in LDS

        const float* as = As[buf];
        const float* bs = Bs[buf];
#pragma unroll
        for (int kk = 0; kk < TBK; kk += 4) {
            const int ko = kk + half * 2;
            // A 16x4 fragments: lanes0-15 K={0,1}, lanes16-31 K={2,3}
            const v2f a0 = *(const v2f*)(as + arow0 + ko);
            const v2f a1 = *(const v2f*)(as + arow1 + ko);
            // B 4x16 fragments: same K split, N striped across 16 lanes
            v2f b0, b1;
            b0.x = bs[ko * BS_STRIDE + cb];
            b0.y = bs[(ko + 1) * BS_STRIDE + cb];
            b1.x = bs[ko * BS_STRIDE + cb + 16];
            b1.y = bs[(ko + 1) * BS_STRIDE + cb + 16];
            acc00 = __builtin_amdgcn_wmma_f32_16x16x4_f32(false, a0, false, b0,
                                                          (short)0, acc00, false, false);
            acc01 = __builtin_amdgcn_wmma_f32_16x16x4_f32(false, a0, false, b1,
                                                          (short)0, acc01, false, false);
            acc10 = __builtin_amdgcn_wmma_f32_16x16x4_f32(false, a1, false, b0,
                                                          (short)0, acc10, false, false);
            acc11 = __builtin_amdgcn_wmma_f32_16x16x4_f32(false, a1, false, b1,
                                                          (short)0, acc11, false, false);
        }
        __syncthreads();                  // done reading buf before it is refilled
        buf ^= 1;
    }

    // C/D layout: VGPR r -> M = r (lanes 0-15) / M = 8+r (lanes 16-31); N = lane&15
    const int row0 = bm + wm * 32 + half * 8;
    const int col0 = bn + wn * 32 + l15;
    const float bv0 = bias[col0];
    const float bv1 = bias[col0 + 16];
#pragma unroll
    for (int r = 0; r < 8; r++) {
        C[(size_t)(row0 + r) * N + col0]           = acc00[r] + bv0;
        C[(size_t)(row0 + r) * N + col0 + 16]      = acc01[r] + bv1;
        C[(size_t)(row0 + 16 + r) * N + col0]      = acc10[r] + bv0;
        C[(size_t)(row0 + 16 + r) * N + col0 + 16] = acc11[r] + bv1;
    }
}

// RoPE in-place on q,k inside qkv[S][NH*3*HD]. One thread per (s, head, pair).
__global__ __launch_bounds__(256)
void rope_kernel(float* __restrict__ qkv, const float* __restrict__ rot) {
    const int idx = blockIdx.x * blockDim.x + threadIdx.x; // S*NH*32
    const int i = idx & 31;
    const int n = (idx >> 5) & 15;
    const int s = idx >> 9;
    if (s >= SEQ) return;

    const float r  = rot[s * 32 + i];
    const float c  = cosf(r);
    const float sn = sinf(r);

    float* base = qkv + (size_t)s * (NH * 3 * HD) + n * (3 * HD);
    {
        float t1 = base[2 * i];
        float t2 = base[2 * i + 1];
        base[2 * i]     = t1 * c - t2 * sn;
        base[2 * i + 1] = t1 * sn + t2 * c;
    }
    {
        float t1 = base[HD + 2 * i];
        float t2 = base[HD + 2 * i + 1];
        base[HD + 2 * i]     = t1 * c - t2 * sn;
        base[HD + 2 * i + 1] = t1 * sn + t2 * c;
    }
}

// Sliding-window attention. One 64-thread block per (s, head).
// Zero-padded k/v semantics: OOB score = 0 (kept in softmax), OOB v contributes 0.
#define MAXW 128
__global__ __launch_bounds__(64)
void attn_kernel(const float* __restrict__ qkv,
                 float* __restrict__ ctx,
                 const int* __restrict__ wptr) {
    const int n = blockIdx.x & (NH - 1);
    const int s = blockIdx.x >> 4;
    const int W = *wptr;
    const int pad = W >> 1;
    const int tid = threadIdx.x;

    __shared__ float qs[HD];
    __shared__ float sc[MAXW];
    __shared__ float red[2];

    const size_t rowstride = NH * 3 * HD; // 3072
    qs[tid] = qkv[(size_t)s * rowstride + n * (3 * HD) + tid];
    __syncthreads();

    for (int j = tid; j < W; j += 64) {
        const int p = s - pad + j;
        float d = 0.0f;
        if (p >= 0 && p < SEQ) {
            const float* kr = qkv + (size_t)p * rowstride + n * (3 * HD) + HD;
#pragma unroll 8
            for (int e = 0; e < HD; e++) d += qs[e] * kr[e];
            d *= 0.125f; // 1/sqrt(64)
        }
        sc[j] = d;
    }
    __syncthreads();

    if (tid == 0) {
        float m = -3.0e38f;
        for (int j = 0; j < W; j++) m = fmaxf(m, sc[j]);
        red[0] = m;
    }
    __syncthreads();
    for (int j = tid; j < W; j += 64) sc[j] = expf(sc[j] - red[0]);
    __syncthreads();
    if (tid == 0) {
        float sum = 0.0f;
        for (int j = 0; j < W; j++) sum += sc[j];
        red[1] = 1.0f / sum;
    }
    __syncthreads();

    float acc = 0.0f;
    for (int j = 0; j < W; j++) {
        const int p = s - pad + j;
        if (p >= 0 && p < SEQ)
            acc += sc[j] * qkv[(size_t)p * rowstride + n * (3 * HD) + 2 * HD + tid];
    }
    ctx[(size_t)s * HID + n * HD + tid] = acc * red[1];
}

extern "C" void kernel_launch(void* const* d_in, const int* in_sizes, int n_in,
                              void* d_out, int out_size, void* d_ws, size_t ws_size,
                              hipStream_t stream) {
    const float* x     = (const float*)d_in[0];  // [4096,1024]
    const float* rot   = (const float*)d_in[1];  // [4096,32]
    const float* W_qkv = (const float*)d_in[2];  // [1024,3072]
    const float* b_qkv = (const float*)d_in[3];  // [3072]
    const float* W_out = (const float*)d_in[4];  // [1024,1024]
    const float* b_out = (const float*)d_in[5];  // [1024]
    const int*   wsz   = (const int*)d_in[6];    // scalar window_size

    float* out = (float*)d_out;                  // [4096,1024]

    float* qkv = (float*)d_ws;                   // [4096,3072]  (48 MB)
    float* ctx = qkv + (size_t)SEQ * (3 * HID);  // [4096,1024]  (16 MB)

    // 1) QKV projection: M=4096, N=3072, K=1024
    {
        dim3 grid(3 * HID / TBN, SEQ / TBM);
        gemm_bias_f32_wmma<<<grid, 256, 0, stream>>>(x, W_qkv, b_qkv, qkv,
                                                     SEQ, 3 * HID, HID);
    }
    // 2) RoPE on q,k
    {
        const int total = SEQ * NH * 32;
        rope_kernel<<<total / 256, 256, 0, stream>>>(qkv, rot);
    }
    // 3) sliding-window attention
    {
        attn_kernel<<<SEQ * NH, 64, 0, stream>>>(qkv, ctx, wsz);
    }
    // 4) output projection: M=4096, N=1024, K=1024
    {
        dim3 grid(HID / TBN, SEQ / TBM);
        gemm_bias_f32_wmma<<<grid, 256, 0, stream>>>(ctx, W_out, b_out, out,
                                                     SEQ, HID, HID);
    }
}